// ProcessContinuous_30786325577969
// MI455X (gfx1250) — compile-verified
//
#include <hip/hip_runtime.h>

typedef __attribute__((ext_vector_type(2))) float v2f;
typedef __attribute__((ext_vector_type(8))) float v8f;

#define NCOL 512              // 4 * D
#define ROWS_PER_WAVE 16
#define WAVES_PER_BLOCK 8
#define BLOCK_THREADS (WAVES_PER_BLOCK * 32)

// out[row, c] = relu( scalar_{c&3}(row) * w_{c&3}[c>>2] + b_{c&3}[c>>2] )
// scalar_0 = pot_odds, scalar_1 = amnt_to_call, scalar_2 = hero_stack, scalar_3 = pot_odds
// Implemented as a K=4 GEMM per 16-row block via V_WMMA_F32_16X16X4_F32.
__global__ __launch_bounds__(BLOCK_THREADS)
void poker_branch_wmma_kernel(
    const float* __restrict__ x,
    const float* __restrict__ stack_w, const float* __restrict__ stack_b,
    const float* __restrict__ call_w,  const float* __restrict__ call_b,
    const float* __restrict__ odds_w,  const float* __restrict__ odds_b,
    const float* __restrict__ pot_w,   const float* __restrict__ pot_b,
    float* __restrict__ out, int n_rows)
{
    // Per-column fused weight/bias tables: the only nonzero of column c in the
    // 4x512 B matrix is W[c&3][c>>2]; bias likewise.
    __shared__ float wCol[NCOL];
    __shared__ float bCol[NCOL];

    for (int i = threadIdx.x; i < NCOL; i += BLOCK_THREADS) {
        const int d = i >> 2, s = i & 3;
        const float* W  = (s == 0) ? pot_w : (s == 1) ? call_w : (s == 2) ? stack_w : odds_w;
        const float* Bv = (s == 0) ? pot_b : (s == 1) ? call_b : (s == 2) ? stack_b : odds_b;
        wCol[i] = W[d];
        bCol[i] = Bv[d];
    }
    __syncthreads();

    const int lane = threadIdx.x & 31;
    const int wave = threadIdx.x >> 5;
    const int rowBase = (blockIdx.x * WAVES_PER_BLOCK + wave) * ROWS_PER_WAVE;
    if (rowBase >= n_rows) return;   // wave-uniform guard (EXEC stays all-ones for WMMA)

    const int m  = lane & 15;        // row within 16-row block / column within tile
    const int hi = lane >> 4;        // lane half selects K rows {0,1} vs {2,3}

    // A matrix 16x4 f32 layout: VGPR0 = K0 (lanes 0-15) / K2 (lanes 16-31),
    //                           VGPR1 = K1 (lanes 0-15) / K3 (lanes 16-31).
    const float* xr = x + (size_t)(rowBase + m) * 16;
    const float s_stack = xr[0];   // hero_stack   (K=2)
    const float s_call  = xr[5];   // amnt_to_call (K=1)
    const float s_odds  = xr[6];   // pot_odds     (K=0 and K=3)

    v2f a;
    a.x = hi ? s_stack : s_odds;   // K=2 : K=0
    a.y = hi ? s_odds  : s_call;   // K=3 : K=1

    const int klo = hi * 2;        // B rows held in b.x / b.y by this lane half
    const int khi = klo + 1;

    // D layout: VGPR r -> row (rowBase + r + hi*8), column c = t*16 + m.
    float* outBase = out + (size_t)(rowBase + hi * 8) * NCOL + m;

    #pragma unroll 4
    for (int t = 0; t < NCOL / 16; ++t) {
        const int   c    = t * 16 + m;
        const int   sel  = c & 3;
        const float wv   = wCol[c];
        const float bias = bCol[c];

        v2f b;
        b.x = (sel == klo) ? wv : 0.0f;
        b.y = (sel == khi) ? wv : 0.0f;

        v8f cacc;
        #pragma unroll
        for (int r = 0; r < 8; ++r) cacc[r] = bias;   // bias independent of row

        // D = A(16x4) x B(4x16) + C ; exact fp32 path
        v8f dres = __builtin_amdgcn_wmma_f32_16x16x4_f32(
            false, a, false, b, (short)0, cacc, false, false);

        float* op = outBase + t * 16;
        #pragma unroll
        for (int r = 0; r < 8; ++r) {
            const float v = fmaxf(dres[r], 0.0f);     // relu
            __builtin_nontemporal_store(v, op + (size_t)r * NCOL);  // 512MB stream -> NT
        }
    }
}

extern "C" void kernel_launch(void* const* d_in, const int* in_sizes, int n_in,
                              void* d_out, int out_size, void* d_ws, size_t ws_size,
                              hipStream_t stream)
{
    const float* x       = (const float*)d_in[0];
    const float* stack_w = (const float*)d_in[1];
    const float* stack_b = (const float*)d_in[2];
    const float* call_w  = (const float*)d_in[3];
    const float* call_b  = (const float*)d_in[4];
    const float* odds_w  = (const float*)d_in[5];
    const float* odds_b  = (const float*)d_in[6];
    const float* pot_w   = (const float*)d_in[7];
    const float* pot_b   = (const float*)d_in[8];
    float* out = (float*)d_out;

    const int n_rows = in_sizes[0] / 16;   // B*M rows of the feature tensor
    const int waves  = (n_rows + ROWS_PER_WAVE - 1) / ROWS_PER_WAVE;
    const int blocks = (waves + WAVES_PER_BLOCK - 1) / WAVES_PER_BLOCK;

    poker_branch_wmma_kernel<<<blocks, BLOCK_THREADS, 0, stream>>>(
        x, stack_w, stack_b, call_w, call_b, odds_w, odds_b, pot_w, pot_b,
        out, n_rows);
}